// VQVAE_65000035058439
// MI455X (gfx1250) — compile-verified
//
#include <hip/hip_runtime.h>
#include <hip/hip_bf16.h>

// ---------------------------------------------------------------------------
// VQ-VAE forward for MI455X (gfx1250), all GEMM-like work on the fp32 WMMA
// path (V_WMMA_F32_16X16X4_F32) to preserve the reference's fp32 numerics.
// Every conv / convT / VQ-distance is an implicit GEMM with 16x16 wave tiles.
// Conv geometry is templated so the K-decomposition is constant-folded
// (no runtime division in the hot loop).
// ---------------------------------------------------------------------------

typedef __attribute__((ext_vector_type(2))) float v2f;
typedef __attribute__((ext_vector_type(8))) float v8f;

#define NBATCH 64

// ---------------------------------------------------------------------------
// Generic implicit-GEMM convolution (NCHW, OIHW weights).
//   out[b][m][oh][ow] = bias[m] + sum_{ci,kh,kw} in[b][ci][ih][iw]*w[m][ci][kh][kw]
// DIL==2 means the INPUT is 2x zero-dilated (transposed conv expressed as a
// dilated-input conv with pad=2 and pre-flipped weights).
// One wave computes a 16(Cout) x 16(pixel) tile; K = Cin*KS*KS stepped by 4
// through V_WMMA_F32_16X16X4_F32. EXEC stays full through the WMMA (bounds
// handled with selects), per ISA 7.12 restrictions.
// A frag: lane M = lane&15, K-pair = (lane>>4)*2.  B frag: col N = lane&15,
// same K-pair.  D frag: VGPR r -> row r (lanes 0-15) / r+8 (lanes 16-31).
// ---------------------------------------------------------------------------
template <int KS, int STRIDE, int PAD, int DIL>
__global__ __launch_bounds__(128)
void conv_wmma(const float* __restrict__ in, const float* __restrict__ wgt,
               const float* __restrict__ bias, const float* __restrict__ skip,
               float* __restrict__ out,
               int Cin, int Cout, int Hin, int Win, int Hout, int Wout,
               int relu)
{
    const int lane = threadIdx.x & 31;
    const int wv   = threadIdx.x >> 5;
    const int half = lane >> 4;       // K-pair select (A/B frag layout)
    const int l15  = lane & 15;
    const int HWo  = Hout * Wout;
    const int npix = NBATCH * HWo;
    const int ntile = blockIdx.x * 4 + wv;
    if (ntile * 16 >= npix) return;           // uniform per wave
    const int mbase = blockIdx.y * 16;

    const int p  = ntile * 16 + l15;          // this lane's output pixel (B col)
    const int bN = p / HWo;                   // npix % 16 == 0 for all layers
    const int s  = p - bN * HWo;
    const int oh = s / Wout;
    const int ow = s - oh * Wout;
    const int HWi = Hin * Win;
    constexpr int KK2 = KS * KS;
    const int K   = Cin * KK2;                // always a multiple of 4 here
    const int m   = mbase + l15;              // this lane's A row (weight row)
    const long wrow = (long)m * K;
    const long ibase = (long)bN * Cin * HWi;

    v8f c = {0.f,0.f,0.f,0.f,0.f,0.f,0.f,0.f};

    for (int k = 0; k < K; k += 4) {
        const int kk0 = k + half * 2;
        v2f a, b;
        #pragma unroll
        for (int j = 0; j < 2; ++j) {
            const int kk  = kk0 + j;
            const int cin = kk / KK2;         // constant divisor -> shifts/magic
            const int rem = kk - cin * KK2;
            const int kh  = rem / KS;
            const int kw  = rem - kh * KS;
            a[j] = wgt[wrow + kk];
            int ihn = oh * STRIDE - PAD + kh;
            int iwn = ow * STRIDE - PAD + kw;
            float bv = 0.f;
            if (DIL == 2) {
                bool ok = (ihn >= 0) && (iwn >= 0) && !(ihn & 1) && !(iwn & 1);
                ihn >>= 1; iwn >>= 1;
                ok = ok && (ihn < Hin) && (iwn < Win);
                bv = ok ? in[ibase + (long)cin * HWi + ihn * Win + iwn] : 0.f;
            } else if (PAD != 0) {
                bool ok = (ihn >= 0) && (ihn < Hin) && (iwn >= 0) && (iwn < Win);
                bv = ok ? in[ibase + (long)cin * HWi + ihn * Win + iwn] : 0.f;
            } else {                          // 1x1, pad 0: always in bounds
                bv = in[ibase + (long)cin * HWi + ihn * Win + iwn];
            }
            b[j] = bv;
        }
        c = __builtin_amdgcn_wmma_f32_16x16x4_f32(false, a, false, b,
                                                  (short)0, c, false, false);
    }

    #pragma unroll
    for (int r = 0; r < 8; ++r) {
        const int  mm    = mbase + r + half * 8;
        const long oaddr = ((long)bN * Cout + mm) * HWo + s;
        float v = c[r] + bias[mm];
        if (skip) v += skip[oaddr];
        if (relu) v = v > 0.f ? v : 0.f;
        out[oaddr] = v;
    }
}

// ---------------------------------------------------------------------------
// VQ argmin: one wave handles 16 tokens x all 512 codes.
// score(m,n) = z_m . e_n via WMMA over K=512; d'(m,n) = |e_n|^2 - 2*score
// (|z|^2 dropped: constant per row). embT is the pre-transposed codebook
// [k][code] so the 16 lanes of each half read consecutive floats (coalesced);
// the whole 1MB embT stays resident in L2 for all 1024 waves.
// Running per-row min folded per code tile, final cross-lane min within
// 16-lane halves, ties -> lowest index (matches jnp.argmin).
// ---------------------------------------------------------------------------
__global__ __launch_bounds__(32)
void vq_argmin(const float* __restrict__ z, const float* __restrict__ embT,
               const float* __restrict__ embn, int* __restrict__ idxout,
               float* __restrict__ counts)
{
    const int lane = threadIdx.x & 31;
    const int half = lane >> 4;
    const int l15  = lane & 15;
    const int tile = blockIdx.x;            // 1024 tiles of 16 tokens
    const int p    = tile * 16 + l15;       // token for A loads (row = l15)
    const int bN   = p >> 8;
    const int s    = p & 255;
    const long zbase = (long)bN * 512 * 256 + s;

    float bestV[8];
    int   bestI[8];
    #pragma unroll
    for (int r = 0; r < 8; ++r) { bestV[r] = 3.4e38f; bestI[r] = 0; }

    for (int ct = 0; ct < 32; ++ct) {
        const int code = ct * 16 + l15;     // this lane's B column
        v8f c = {0.f,0.f,0.f,0.f,0.f,0.f,0.f,0.f};
        for (int k = 0; k < 512; k += 4) {
            const int kk0 = k + half * 2;
            v2f a, b;
            a[0] = z[zbase + (long)(kk0    ) * 256];
            a[1] = z[zbase + (long)(kk0 + 1) * 256];
            b[0] = embT[(long)(kk0    ) * 512 + code];
            b[1] = embT[(long)(kk0 + 1) * 512 + code];
            c = __builtin_amdgcn_wmma_f32_16x16x4_f32(false, a, false, b,
                                                      (short)0, c, false, false);
        }
        const float en = embn[code];
        #pragma unroll
        for (int r = 0; r < 8; ++r) {
            const float d = en - 2.0f * c[r];
            if (d < bestV[r] || (d == bestV[r] && code < bestI[r])) {
                bestV[r] = d; bestI[r] = code;
            }
        }
    }

    #pragma unroll
    for (int r = 0; r < 8; ++r) {
        float v = bestV[r];
        int   i = bestI[r];
        #pragma unroll
        for (int off = 8; off > 0; off >>= 1) {     // stays within 16-lane half
            const float ov = __shfl_xor(v, off, 32);
            const int   oi = __shfl_xor(i, off, 32);
            if (ov < v || (ov == v && oi < i)) { v = ov; i = oi; }
        }
        if (l15 == 0) {
            const int token = tile * 16 + r + half * 8;  // D rows: r / r+8
            idxout[token] = i;
            atomicAdd(&counts[i], 1.0f);
        }
    }
}

// Gather q = embedding[idx] into NCHW layout; accumulate sum((q-z)^2).
// Straight-through: forward value of q_st is exactly q.
__global__ __launch_bounds__(256)
void vq_gather(const int* __restrict__ idx, const float* __restrict__ emb,
               const float* __restrict__ z, float* __restrict__ q,
               float* __restrict__ vq_sum)
{
    __shared__ float red[256];
    const long t = (long)blockIdx.x * 256 + threadIdx.x;  // 16384*512 exact
    const int  p = (int)(t >> 9);
    const int  d = (int)(t & 511);
    const int  bN = p >> 8;
    const int  s  = p & 255;
    const float qv = emb[(long)idx[p] * 512 + d];
    const long zaddr = ((long)bN * 512 + d) * 256 + s;
    const float df = qv - z[zaddr];
    q[zaddr] = qv;
    red[threadIdx.x] = df * df;
    __syncthreads();
    for (int o = 128; o > 0; o >>= 1) {
        if (threadIdx.x < o) red[threadIdx.x] += red[threadIdx.x + o];
        __syncthreads();
    }
    if (threadIdx.x == 0) atomicAdd(vq_sum, red[0]);
}

// ---------------------------------------------------------------------------
// Small helper kernels
// ---------------------------------------------------------------------------
__global__ void norm_k(const float* __restrict__ img, float* __restrict__ xn)
{
    const int t = blockIdx.x * 256 + threadIdx.x;
    xn[t] = 2.0f * (img[t] - 0.5f);
}

// per-code squared norm + transpose of the codebook: embT[k][code] = emb[code][k]
__global__ void embprep_k(const float* __restrict__ emb, float* __restrict__ embn,
                          float* __restrict__ embT)
{
    const int code = blockIdx.x;            // 512 blocks of 256 threads
    float acc = 0.f;
    const long row = (long)code * 512;
    for (int d = threadIdx.x; d < 512; d += 256) {
        const float v = emb[row + d];
        embT[(long)d * 512 + code] = v;
        acc += v * v;
    }
    __shared__ float red[256];
    red[threadIdx.x] = acc;
    __syncthreads();
    for (int o = 128; o > 0; o >>= 1) {
        if (threadIdx.x < o) red[threadIdx.x] += red[threadIdx.x + o];
        __syncthreads();
    }
    if (threadIdx.x == 0) embn[code] = red[0];
}

// wf[o][i][kh][kw] = w[i][o][3-kh][3-kw]   (ConvTranspose -> conv weights)
__global__ void repackT_k(const float* __restrict__ w, float* __restrict__ wf,
                          int Cin, int Cout)
{
    const int t = blockIdx.x * 256 + threadIdx.x;
    const int total = Cout * Cin * 16;
    if (t >= total) return;
    const int o   = t / (Cin * 16);
    const int rem = t - o * (Cin * 16);
    const int i   = rem >> 4;
    const int kk  = rem & 15;
    const int kh  = kk >> 2, kw = kk & 3;
    wf[t] = w[(((long)i * Cout + o) * 4 + (3 - kh)) * 4 + (3 - kw)];
}

// dec_out: 1x1 conv 16 -> 1 channel
__global__ void dec_out_k(const float* __restrict__ in, const float* __restrict__ w,
                          const float* __restrict__ b, float* __restrict__ out)
{
    const int t = blockIdx.x * 256 + threadIdx.x;   // NBATCH*65536 pixels
    const int bN = t >> 16;
    const int s  = t & 65535;
    const long base = (long)bN * 16 * 65536 + s;
    float acc = b[0];
    #pragma unroll
    for (int c2 = 0; c2 < 16; ++c2) acc += in[base + (long)c2 * 65536] * w[c2];
    out[t] = acc;
}

__global__ __launch_bounds__(256)
void reduce_sq_k(const float* __restrict__ a, const float* __restrict__ b,
                 long n, float* __restrict__ out)
{
    __shared__ float red[256];
    float acc = 0.f;
    for (long i = (long)blockIdx.x * 256 + threadIdx.x; i < n;
         i += (long)gridDim.x * 256) {
        const float d = a[i] - b[i];
        acc += d * d;
    }
    red[threadIdx.x] = acc;
    __syncthreads();
    for (int o = 128; o > 0; o >>= 1) {
        if (threadIdx.x < o) red[threadIdx.x] += red[threadIdx.x + o];
        __syncthreads();
    }
    if (threadIdx.x == 0) atomicAdd(out, red[0]);
}

__global__ __launch_bounds__(512)
void perplexity_k(const float* __restrict__ counts, float* __restrict__ outp)
{
    __shared__ float red[512];
    const int t = threadIdx.x;
    const float avg = counts[t] * (1.0f / 16384.0f);
    red[t] = avg * __logf(avg + 1e-10f);
    __syncthreads();
    for (int o = 256; o > 0; o >>= 1) {
        if (t < o) red[t] += red[t + o];
        __syncthreads();
    }
    if (t == 0) outp[0] = __expf(-red[0]);
}

__global__ void combine_k(const float* __restrict__ scal, float* __restrict__ loss)
{
    // scal[0] = sum((q-z)^2) over 8388608; scal[1] = sum((out-label)^2) over 4194304
    loss[0] = scal[1] * (1.0f / 4194304.0f) + 0.25f * scal[0] * (1.0f / 8388608.0f);
}

// ---------------------------------------------------------------------------
// Host orchestration
// ---------------------------------------------------------------------------
extern "C" void kernel_launch(void* const* d_in, const int* in_sizes, int n_in,
                              void* d_out, int out_size, void* d_ws, size_t ws_size,
                              hipStream_t stream)
{
    // ---- input pointers (setup_inputs dict flattened in insertion order) ----
    const float* img  = (const float*)d_in[0];
    const float* ec_w[4] = {(const float*)d_in[1], (const float*)d_in[3],
                            (const float*)d_in[5], (const float*)d_in[7]};
    const float* ec_b[4] = {(const float*)d_in[2], (const float*)d_in[4],
                            (const float*)d_in[6], (const float*)d_in[8]};
    // enc_res[0]: 9..14, enc_res[1]: 15..20  (w1,b1,w2,b2,w3,b3)
    const float* enc_out_w = (const float*)d_in[21];
    const float* enc_out_b = (const float*)d_in[22];
    const float* dec_in_w  = (const float*)d_in[23];
    const float* dec_in_b  = (const float*)d_in[24];
    // dec_res[0]: 25..30, dec_res[1]: 31..36
    const float* tw[4] = {(const float*)d_in[37], (const float*)d_in[39],
                          (const float*)d_in[41], (const float*)d_in[43]};
    const float* tb[4] = {(const float*)d_in[38], (const float*)d_in[40],
                          (const float*)d_in[42], (const float*)d_in[44]};
    const float* dec_out_w = (const float*)d_in[45];
    const float* dec_out_b = (const float*)d_in[46];
    const float* emb       = (const float*)d_in[47];

    float* outp = (float*)d_out;           // [0]=loss, [1..4194304]=out, [last]=perplexity
    float* outimg = outp + 1;

    // ---- workspace layout (floats); total ~624 MB ----
    float* F = (float*)d_ws;
    size_t off = 0;
    auto alloc = [&](size_t n) { float* p = F + off; off += n; return p; };
    float* XN = alloc(4194304);     // normalized input / label
    float* A  = alloc(16777216);
    float* Bb = alloc(8388608);
    float* S0 = alloc(2097152);
    float* S1 = alloc(2097152);
    float* S2 = alloc(2097152);
    float* S3 = alloc(2097152);
    float* Z  = alloc(8388608);     // z, later reused for convT1 output
    float* Q  = alloc(8388608);
    float* E3 = alloc(33554432);
    float* E4 = alloc(67108864);
    float* WT1 = alloc(128 * 128 * 16);
    float* WT2 = alloc(128 * 64 * 16);
    float* WT3 = alloc(64 * 32 * 16);
    float* WT4 = alloc(32 * 16 * 16);
    float* EMBT = alloc(512 * 512); // transposed codebook [k][code]
    float* EMBN = alloc(512);
    int*   IDX  = (int*)alloc(16384);
    float* CNT  = alloc(512);
    float* SCAL = alloc(8);         // [0]=vq sum, [1]=recon sum

    hipMemsetAsync(CNT, 0, 512 * sizeof(float), stream);
    hipMemsetAsync(SCAL, 0, 8 * sizeof(float), stream);

    auto conv = [&](const float* in, const float* w, const float* bias,
                    const float* skip, float* out, int Cin, int Cout,
                    int Hin, int Win, int ks, int stride, int pad,
                    int dil, int relu) {
        const int Hout = (dil == 2) ? Hin * 2 : (Hin + 2 * pad - ks) / stride + 1;
        const int Wout = (dil == 2) ? Win * 2 : (Win + 2 * pad - ks) / stride + 1;
        const int npix = NBATCH * Hout * Wout;
        const int ntiles = npix / 16;
        dim3 grid((ntiles + 3) / 4, Cout / 16);
        if (dil == 2)
            conv_wmma<4, 1, 2, 2><<<grid, 128, 0, stream>>>(in, w, bias, skip, out,
                Cin, Cout, Hin, Win, Hout, Wout, relu);
        else if (ks == 4)
            conv_wmma<4, 2, 1, 1><<<grid, 128, 0, stream>>>(in, w, bias, skip, out,
                Cin, Cout, Hin, Win, Hout, Wout, relu);
        else if (ks == 3)
            conv_wmma<3, 1, 1, 1><<<grid, 128, 0, stream>>>(in, w, bias, skip, out,
                Cin, Cout, Hin, Win, Hout, Wout, relu);
        else
            conv_wmma<1, 1, 0, 1><<<grid, 128, 0, stream>>>(in, w, bias, skip, out,
                Cin, Cout, Hin, Win, Hout, Wout, relu);
    };
    auto resblock = [&](const float* x, float* h1, float* h2, float* r,
                        int base /* d_in index of w1 */) {
        const float* w1 = (const float*)d_in[base + 0];
        const float* b1 = (const float*)d_in[base + 1];
        const float* w2 = (const float*)d_in[base + 2];
        const float* b2 = (const float*)d_in[base + 3];
        const float* w3 = (const float*)d_in[base + 4];
        const float* b3 = (const float*)d_in[base + 5];
        conv(x,  w1, b1, nullptr, h1, 128, 128, 16, 16, 3, 1, 1, 1, 1);
        conv(h1, w2, b2, nullptr, h2, 128, 128, 16, 16, 3, 1, 1, 1, 1);
        conv(h2, w3, b3, x,       r,  128, 128, 16, 16, 1, 1, 0, 1, 0);
    };

    // ---- prep ----
    norm_k<<<16384, 256, 0, stream>>>(img, XN);
    embprep_k<<<512, 256, 0, stream>>>(emb, EMBN, EMBT);
    repackT_k<<<(128 * 128 * 16 + 255) / 256, 256, 0, stream>>>(tw[0], WT1, 128, 128);
    repackT_k<<<(128 * 64 * 16 + 255) / 256, 256, 0, stream>>>(tw[1], WT2, 128, 64);
    repackT_k<<<(64 * 32 * 16 + 255) / 256, 256, 0, stream>>>(tw[2], WT3, 64, 32);
    repackT_k<<<(32 * 16 * 16 + 255) / 256, 256, 0, stream>>>(tw[3], WT4, 32, 16);

    // ---- encoder ----
    conv(XN, ec_w[0], ec_b[0], nullptr, A,  1,   16, 256, 256, 4, 2, 1, 1, 1);
    conv(A,  ec_w[1], ec_b[1], nullptr, Bb, 16,  32, 128, 128, 4, 2, 1, 1, 1);
    conv(Bb, ec_w[2], ec_b[2], nullptr, A,  32,  64, 64,  64,  4, 2, 1, 1, 1);
    conv(A,  ec_w[3], ec_b[3], nullptr, S0, 64, 128, 32,  32,  4, 2, 1, 1, 1);
    resblock(S0, S1, S2, S3, 9);
    resblock(S3, S1, S2, S0, 15);
    conv(S0, enc_out_w, enc_out_b, nullptr, Z, 128, 512, 16, 16, 1, 1, 0, 1, 0);

    // ---- vector quantizer ----
    vq_argmin<<<1024, 32, 0, stream>>>(Z, EMBT, EMBN, IDX, CNT);
    vq_gather<<<32768, 256, 0, stream>>>(IDX, emb, Z, Q, &SCAL[0]);

    // ---- decoder ----
    conv(Q, dec_in_w, dec_in_b, nullptr, S1, 512, 128, 16, 16, 1, 1, 0, 1, 1);
    resblock(S1, S2, S3, S0, 25);
    resblock(S0, S1, S2, S3, 31);
    conv(S3, WT1, tb[0], nullptr, Z,  128, 128, 16,  16,  4, 1, 2, 2, 1);  // -> 32x32
    conv(Z,  WT2, tb[1], nullptr, A,  128, 64,  32,  32,  4, 1, 2, 2, 1);  // -> 64x64
    conv(A,  WT3, tb[2], nullptr, E3, 64,  32,  64,  64,  4, 1, 2, 2, 1);  // -> 128x128
    conv(E3, WT4, tb[3], nullptr, E4, 32,  16,  128, 128, 4, 1, 2, 2, 1);  // -> 256x256
    dec_out_k<<<16384, 256, 0, stream>>>(E4, dec_out_w, dec_out_b, outimg);

    // ---- losses & perplexity ----
    reduce_sq_k<<<2048, 256, 0, stream>>>(outimg, XN, (long)4194304, &SCAL[1]);
    perplexity_k<<<1, 512, 0, stream>>>(CNT, outp + 4194305);
    combine_k<<<1, 1, 0, stream>>>(SCAL, outp);
}